// MetaTransformer_54090818126554
// MI455X (gfx1250) — compile-verified
//
#include <hip/hip_runtime.h>

// ---------------------------------------------------------------------------
// MetaTransformer for MI455X (gfx1250), fp32 WMMA (v_wmma_f32_16x16x4_f32).
//
// Math per layer:  G[b] = Z[b][0:2047]^T @ Z[b][0:2047]      (Gram, per batch)
//                  W[b] = (1/N) * sum_h Pfull_h @ G[b] @ Qfull_h^T
//                  Z[b] = Z[b] + Z[b] @ W[b]^T
// Channels padded 257 -> 288 = 18*16 so every GEMM tiles cleanly and the
// N-tile count is even (each wave computes a 16x32 strip = 2 accumulators,
// sharing the A fragment between both WMMAs).
// ---------------------------------------------------------------------------

#define L_    4
#define H_    8
#define D_    256
#define D1    257     // d+1 channels
#define DP    288     // padded channels = 18*16
#define TI    18      // DP/16
#define TH    (TI/2)  // tile pairs along N = 9
#define B_    8
#define N1_   2048
#define NCTX  2047    // tokens entering the Gram (row N excluded)

typedef float v2f __attribute__((ext_vector_type(2)));
typedef float v8f __attribute__((ext_vector_type(8)));

__device__ __forceinline__ v8f wmma_f32(v2f a, v2f b, v8f c) {
  // D = A(16x4,f32) * B(4x16,f32) + C(16x16,f32)
  return __builtin_amdgcn_wmma_f32_16x16x4_f32(
      /*neg_a=*/false, a, /*neg_b=*/false, b,
      /*c_mod=*/(short)0, c, /*reuse_a=*/false, /*reuse_b=*/false);
}

// ---------------------------------------------------------------------------
// Prep: pad Z (B,2048,257) -> Zp (B,2048,288), zero tail columns.
__global__ void pad_z_kernel(const float* __restrict__ Z, float* __restrict__ Zp) {
  int idx = blockIdx.x * blockDim.x + threadIdx.x;
  if (idx >= B_ * N1_ * DP) return;
  int c = idx % DP;
  int bn = idx / DP;
  Zp[idx] = (c < D1) ? Z[(size_t)bn * D1 + c] : 0.0f;
}

// Prep: build padded P_full / Q_full (L,H,288,288) from allparam (L,H,2,256,256).
// Pfull[i][k] = P[i][k] (i,k<256); Pfull[256][256] = 1; else 0.
// Qfull[k][j] = Q[k][j] (k,j<256); else 0.
__global__ void pad_params_kernel(const float* __restrict__ ap,
                                  float* __restrict__ Pp, float* __restrict__ Qp) {
  int idx = blockIdx.x * blockDim.x + threadIdx.x;
  if (idx >= L_ * H_ * DP * DP) return;
  int k = idx % DP;
  int rest = idx / DP;
  int i = rest % DP;
  int lh = rest / DP;
  float p = 0.0f, q = 0.0f;
  if (i < D_ && k < D_) {
    const float* base = ap + (size_t)lh * 2 * D_ * D_;
    p = base[(size_t)i * D_ + k];
    q = base[(size_t)D_ * D_ + (size_t)i * D_ + k];
  } else if (i == D_ && k == D_) {
    p = 1.0f;
  }
  Pp[idx] = p;
  Qp[idx] = q;
}

// ---------------------------------------------------------------------------
// G[b] = Zp[b][0:2047]^T @ Zp[b][0:2047]   (DPxDP, K over 2047 tokens)
// Each wave: 16x32 output strip (tiles j0 and j0+16 share the A fragment).
// Last K-group (tokens 2044..2047, with 2047 zeroed) peeled out of the loop
// so the hot loop has no lane-divergent conditionals.
__global__ void gram_kernel(const float* __restrict__ Zp, float* __restrict__ G) {
  const int wave = threadIdx.x >> 5, lane = threadIdx.x & 31;
  const int task = blockIdx.x * (blockDim.x >> 5) + wave;
  if (task >= TI * TH) return;
  const int b = blockIdx.y;
  const int i0 = (task / TH) * 16, j0 = (task % TH) * 32;
  const int half = lane >> 4, l16 = lane & 15;
  const float* Zb = Zp + (size_t)b * N1_ * DP;
  const float* colA  = Zb + i0 + l16;
  const float* colB0 = Zb + j0 + l16;
  const float* colB1 = Zb + j0 + 16 + l16;
  v8f acc0 = {}, acc1 = {};
#pragma unroll 4
  for (int m = 0; m < 2044; m += 4) {          // 511 full K-groups
    const size_t r0 = (size_t)(m + 2 * half) * DP;
    const size_t r1 = r0 + DP;
    v2f a, b0, b1;
    a.x  = colA[r0];  a.y  = colA[r1];
    b0.x = colB0[r0]; b0.y = colB0[r1];
    b1.x = colB1[r0]; b1.y = colB1[r1];
    acc0 = wmma_f32(a, b0, acc0);
    acc1 = wmma_f32(a, b1, acc1);
  }
  {                                            // peeled tail: tokens 2044..2046
    const int rr0 = 2044 + 2 * half;           // 2044 or 2046 (always valid)
    const size_t r0 = (size_t)rr0 * DP;
    const size_t r1 = r0 + DP;
    const bool ok = (rr0 + 1) < NCTX;          // token 2047 contributes zero
    v2f a, b0, b1;
    a.x  = colA[r0];  a.y  = ok ? colA[r1]  : 0.0f;
    b0.x = colB0[r0]; b0.y = ok ? colB0[r1] : 0.0f;
    b1.x = colB1[r0]; b1.y = ok ? colB1[r1] : 0.0f;
    acc0 = wmma_f32(a, b0, acc0);
    acc1 = wmma_f32(a, b1, acc1);
  }
  float* Gb = G + (size_t)b * DP * DP;
  const int row0 = i0 + 8 * half;
#pragma unroll
  for (int r = 0; r < 8; ++r) {
    Gb[(size_t)(row0 + r) * DP + j0 + l16]      = acc0[r];
    Gb[(size_t)(row0 + r) * DP + j0 + 16 + l16] = acc1[r];
  }
}

// ---------------------------------------------------------------------------
// T[b,h] = G[b] @ Qfull_h^T :  T[i][k] = sum_j G[i][j] * Qfull[k][j]
// Each wave: 16x32 strip over k (shares A = G rows between the two WMMAs).
__global__ void t_kernel(const float* __restrict__ G, const float* __restrict__ Qp_l,
                         float* __restrict__ T) {
  const int wave = threadIdx.x >> 5, lane = threadIdx.x & 31;
  const int task = blockIdx.x * (blockDim.x >> 5) + wave;
  if (task >= TI * TH) return;
  const int bh = blockIdx.y;                   // b*8 + h
  const int b = bh >> 3, h = bh & 7;
  const int i0 = (task / TH) * 16, k0 = (task % TH) * 32;
  const int half = lane >> 4, l16 = lane & 15;
  const float* Ga  = G + (size_t)b * DP * DP + (size_t)(i0 + l16) * DP;
  const float* Qh0 = Qp_l + (size_t)h * DP * DP + (size_t)(k0 + l16) * DP;
  const float* Qh1 = Qh0 + (size_t)16 * DP;
  v8f acc0 = {}, acc1 = {};
#pragma unroll 4
  for (int j = 0; j < DP; j += 4) {
    const int jc = j + 2 * half;
    v2f a  = *(const v2f*)(Ga + jc);
    v2f b0 = *(const v2f*)(Qh0 + jc);
    v2f b1 = *(const v2f*)(Qh1 + jc);
    acc0 = wmma_f32(a, b0, acc0);
    acc1 = wmma_f32(a, b1, acc1);
  }
  float* Tt = T + (size_t)bh * DP * DP;
  const int row0 = i0 + 8 * half;
#pragma unroll
  for (int r = 0; r < 8; ++r) {
    Tt[(size_t)(row0 + r) * DP + k0 + l16]      = acc0[r];
    Tt[(size_t)(row0 + r) * DP + k0 + 16 + l16] = acc1[r];
  }
}

// ---------------------------------------------------------------------------
// W[b] = (1/N) * sum_h Pfull_h @ T[b,h]   (16x32 strip per wave, A shared)
__global__ void w_kernel(const float* __restrict__ Pp_l, const float* __restrict__ T,
                         float* __restrict__ W) {
  const int wave = threadIdx.x >> 5, lane = threadIdx.x & 31;
  const int task = blockIdx.x * (blockDim.x >> 5) + wave;
  if (task >= TI * TH) return;
  const int b = blockIdx.y;
  const int i0 = (task / TH) * 16, p0 = (task % TH) * 32;
  const int half = lane >> 4, l16 = lane & 15;
  v8f acc0 = {}, acc1 = {};
  for (int h = 0; h < H_; ++h) {
    const float* Pa = Pp_l + (size_t)h * DP * DP + (size_t)(i0 + l16) * DP;
    const float* Tt = T + (size_t)(b * H_ + h) * DP * DP;
#pragma unroll 4
    for (int k = 0; k < DP; k += 4) {
      const int kc = k + 2 * half;
      v2f a = *(const v2f*)(Pa + kc);
      v2f b0, b1;
      b0.x = Tt[(size_t)kc * DP + p0 + l16];
      b0.y = Tt[(size_t)(kc + 1) * DP + p0 + l16];
      b1.x = Tt[(size_t)kc * DP + p0 + 16 + l16];
      b1.y = Tt[(size_t)(kc + 1) * DP + p0 + 16 + l16];
      acc0 = wmma_f32(a, b0, acc0);
      acc1 = wmma_f32(a, b1, acc1);
    }
  }
  const float scale = 1.0f / (float)NCTX;
  float* Wb = W + (size_t)b * DP * DP;
  const int row0 = i0 + 8 * half;
#pragma unroll
  for (int r = 0; r < 8; ++r) {
    Wb[(size_t)(row0 + r) * DP + p0 + l16]      = acc0[r] * scale;
    Wb[(size_t)(row0 + r) * DP + p0 + 16 + l16] = acc1[r] * scale;
  }
}

// ---------------------------------------------------------------------------
// Zdst[b] = Zsrc[b] + Zsrc[b] @ W[b]^T
// out[n][i] = Z[n][i] + sum_p Z[n][p]*W[i][p]; 16x32 strip over i per wave.
__global__ void update_kernel(const float* __restrict__ Zsrc, const float* __restrict__ W,
                              float* __restrict__ Zdst) {
  const int wave = threadIdx.x >> 5, lane = threadIdx.x & 31;
  const int task = blockIdx.x * (blockDim.x >> 5) + wave;
  const int NT = (N1_ / 16) * TH;              // 128*9 = 1152 strips per batch
  if (task >= NT) return;
  const int b = blockIdx.y;
  const int n0 = (task / TH) * 16, i0 = (task % TH) * 32;
  const int half = lane >> 4, l16 = lane & 15;
  const float* Zb = Zsrc + (size_t)b * N1_ * DP;
  const float* Za = Zb + (size_t)(n0 + l16) * DP;
  const float* W0 = W + (size_t)b * DP * DP + (size_t)(i0 + l16) * DP;
  const float* W1 = W0 + (size_t)16 * DP;
  v8f acc0 = {}, acc1 = {};
#pragma unroll 4
  for (int p = 0; p < DP; p += 4) {
    const int pc = p + 2 * half;
    v2f a  = *(const v2f*)(Za + pc);           // A[M=n][K=p] = Z[n][p]
    v2f b0 = *(const v2f*)(W0 + pc);           // B[K=p][N=i] = W[i][p]
    v2f b1 = *(const v2f*)(W1 + pc);
    acc0 = wmma_f32(a, b0, acc0);
    acc1 = wmma_f32(a, b1, acc1);
  }
  float* Zd = Zdst + (size_t)b * N1_ * DP;
  const int row0 = n0 + 8 * half;
#pragma unroll
  for (int r = 0; r < 8; ++r) {
    const size_t o0 = (size_t)(row0 + r) * DP + i0 + l16;
    const size_t o1 = o0 + 16;
    Zd[o0] = Zb[o0] + acc0[r];
    Zd[o1] = Zb[o1] + acc1[r];
  }
}

// ---------------------------------------------------------------------------
__global__ void unpad_z_kernel(const float* __restrict__ Zp, float* __restrict__ out) {
  int idx = blockIdx.x * blockDim.x + threadIdx.x;
  if (idx >= B_ * N1_ * D1) return;
  int c = idx % D1;
  int bn = idx / D1;
  out[idx] = Zp[(size_t)bn * DP + c];
}

// ---------------------------------------------------------------------------
extern "C" void kernel_launch(void* const* d_in, const int* in_sizes, int n_in,
                              void* d_out, int out_size, void* d_ws, size_t ws_size,
                              hipStream_t stream) {
  const float* Z  = (const float*)d_in[0];   // (8, 2048, 257) f32
  const float* ap = (const float*)d_in[1];   // (4, 8, 2, 256, 256) f32

  float* ws = (float*)d_ws;
  const size_t SZ = (size_t)B_ * N1_ * DP;        // padded Z buffer
  const size_t SG = (size_t)B_ * DP * DP;         // Gram / W
  const size_t ST = (size_t)B_ * H_ * DP * DP;    // per-head temporaries
  const size_t SP = (size_t)L_ * H_ * DP * DP;    // padded params
  float* Z0 = ws;
  float* Z1 = Z0 + SZ;
  float* G  = Z1 + SZ;
  float* T  = G + SG;
  float* W  = T + ST;
  float* Pp = W + SG;
  float* Qp = Pp + SP;                            // total ~85.5 MB

  {
    int n = B_ * N1_ * DP;
    pad_z_kernel<<<(n + 255) / 256, 256, 0, stream>>>(Z, Z0);
  }
  {
    int n = L_ * H_ * DP * DP;
    pad_params_kernel<<<(n + 255) / 256, 256, 0, stream>>>(ap, Pp, Qp);
  }

  float* cur = Z0;
  float* nxt = Z1;
  const dim3 blk(128);                             // 4 waves, one 16x32 strip each
  const int NT_SQ = TI * TH;                       // 162 strips for DPxDP outputs
  const int NT_UP = (N1_ / 16) * TH;               // 1152 strips for Z update
  for (int l = 0; l < L_; ++l) {
    gram_kernel<<<dim3((NT_SQ + 3) / 4, B_), blk, 0, stream>>>(cur, G);
    t_kernel<<<dim3((NT_SQ + 3) / 4, B_ * H_), blk, 0, stream>>>(
        G, Qp + (size_t)l * H_ * DP * DP, T);
    w_kernel<<<dim3((NT_SQ + 3) / 4, B_), blk, 0, stream>>>(
        Pp + (size_t)l * H_ * DP * DP, T, W);
    update_kernel<<<dim3((NT_UP + 3) / 4, B_), blk, 0, stream>>>(cur, W, nxt);
    float* tmp = cur; cur = nxt; nxt = tmp;
  }

  {
    int n = B_ * N1_ * D1;
    unpad_z_kernel<<<(n + 255) / 256, 256, 0, stream>>>(cur, (float*)d_out);
  }
}